// Invert1_10_20684562497570
// MI455X (gfx1250) — compile-verified
//
#include <hip/hip_runtime.h>

typedef float v4f __attribute__((ext_vector_type(4)));
typedef float v2f __attribute__((ext_vector_type(2)));

// ---------------------------------------------------------------------------
// Main kernel: forward spiking recurrence, retimed for packed-f32 FMA.
//   z_t = ((v-T)/(|v|+1) > 0) == (v > T)     (denominator strictly positive;
//                                             v-T of ~unit-magnitude f32s is
//                                             never denormal; exact ties give
//                                             z=0 in both formulations)
// Retimed (z1 = (|x| > T1)):
//   {o,v} <- v_pk_fma_f32(z_t bcast via op_sel, {d_t, -h_{t+1}}, {o,v})
//   z_{t+1} = (v > T_{t+1})
//   o <- fma(z_16, d_16, o);  result = o * sign(x)
// 16 elements/thread: four block-strided b128 NT loads/stores, exact blocks
// of 1024 float4s (4096 floats), ZERO bounds checks. Remainder -> tail.
// ---------------------------------------------------------------------------
__global__ __launch_bounds__(256) void spike_fwd_pk16(
    const float* __restrict__ x,
    const float* __restrict__ cw,   // packed coefs (see pack_coefs)
    float* __restrict__ out)
{
    // Uniform loads -> SGPRs (backend hoists pairs to VGPR pairs once).
    const v2f* __restrict__ cw2 = (const v2f*)cw;
    v2f C[15];
#pragma unroll
    for (int t = 0; t < 15; ++t) C[t] = cw2[t];
    float dlast = cw[30];
    float Ts[16];
#pragma unroll
    for (int t = 0; t < 16; ++t) Ts[t] = cw[32 + t];

    int base = blockIdx.x * 1024 + threadIdx.x;   // float4 index of chunk 0

    const v4f* __restrict__ x4 = (const v4f*)x;
    v4f* __restrict__ o4 = (v4f*)out;

    v4f xv[4];
#pragma unroll
    for (int c = 0; c < 4; ++c) {
        xv[c] = __builtin_nontemporal_load(x4 + base + c * 256);
    }

    float xi[16] = { xv[0].x, xv[0].y, xv[0].z, xv[0].w,
                     xv[1].x, xv[1].y, xv[1].z, xv[1].w,
                     xv[2].x, xv[2].y, xv[2].z, xv[2].w,
                     xv[3].x, xv[3].y, xv[3].z, xv[3].w };
    float res[16];

#pragma unroll
    for (int e = 0; e < 16; ++e) {
        float xe = xi[e];
        float va = __builtin_fabsf(xe);
        // peeled step 1: v1 = |x|, z1 = (v1 > T1)  (abs folds into the cmp)
        float u = (va > Ts[0]) ? 1.0f : 0.0f;
        v2f st;
        st.x = 0.0f;   // o
        st.y = va;     // v
#pragma unroll
        for (int t = 0; t < 15; ++t) {
            v2f uu = { u, u };   // broadcast -> op_sel on v_pk_fma_f32
            st = __builtin_elementwise_fma(uu, C[t], st);
            u = (st.y > Ts[t + 1]) ? 1.0f : 0.0f;
        }
        float o = __builtin_fmaf(u, dlast, st.x);

        // result = o * sign(x): flip sign bit, zero when x == 0
        unsigned sb = __builtin_bit_cast(unsigned, xe) & 0x80000000u;
        float r = __builtin_bit_cast(float, __builtin_bit_cast(unsigned, o) ^ sb);
        res[e] = (xe == 0.0f) ? 0.0f : r;
    }

#pragma unroll
    for (int c = 0; c < 4; ++c) {
        v4f ov;
        ov.x = res[4 * c + 0];
        ov.y = res[4 * c + 1];
        ov.z = res[4 * c + 2];
        ov.w = res[4 * c + 3];
        __builtin_nontemporal_store(ov, o4 + base + c * 256);
    }
}

// ---------------------------------------------------------------------------
// Setup: pack retimed coefficients into workspace so the main kernel gets
// even-aligned 64-bit pairs for v_pk_fma_f32.
//   cw[2t]   = d[t]        (t = 0..14)   out-update coef of step t
//   cw[2t+1] = -h[t+1]                   v-update coef of step t+1
//   cw[30]   = d[15]                     final scalar fma coef
//   cw[32+t] = T[t]        (t = 0..15)   thresholds
// ---------------------------------------------------------------------------
__global__ void pack_coefs(const float* __restrict__ h,
                           const float* __restrict__ d,
                           const float* __restrict__ T,
                           float* __restrict__ cw)
{
    int t = threadIdx.x;
    if (t < 15) {
        cw[2 * t]     = d[t];
        cw[2 * t + 1] = -h[t + 1];
    }
    if (t == 15) {
        cw[30] = d[15];
        cw[31] = 0.0f;
    }
    if (t < 16) cw[32 + t] = T[t];
}

// ---------------------------------------------------------------------------
// Generic scalar path: handles [base, n) elements (remainder of the main
// kernel, or everything in the no-workspace fallback).
// ---------------------------------------------------------------------------
__global__ __launch_bounds__(256) void spike_fwd_tail(
    const float* __restrict__ x,
    const float* __restrict__ h,
    const float* __restrict__ d,
    const float* __restrict__ T,
    float* __restrict__ out,
    int base, int n)
{
    float hs[16], ds[16], Ts[16];
#pragma unroll
    for (int t = 0; t < 16; ++t) { hs[t] = h[t]; ds[t] = d[t]; Ts[t] = T[t]; }

    for (int i = base + blockIdx.x * blockDim.x + threadIdx.x; i < n;
         i += gridDim.x * blockDim.x) {
        float xe = x[i];
        float v  = __builtin_fabsf(xe);
        float z  = 0.0f;
        float o  = 0.0f;
#pragma unroll
        for (int t = 0; t < 16; ++t) {
            v = __builtin_fmaf(z, -hs[t], v);
            z = (v > Ts[t]) ? 1.0f : 0.0f;
            o = __builtin_fmaf(z, ds[t], o);
        }
        unsigned sb = __builtin_bit_cast(unsigned, xe) & 0x80000000u;
        float r = __builtin_bit_cast(float, __builtin_bit_cast(unsigned, o) ^ sb);
        out[i] = (xe == 0.0f) ? 0.0f : r;
    }
}

extern "C" void kernel_launch(void* const* d_in, const int* in_sizes, int n_in,
                              void* d_out, int out_size, void* d_ws, size_t ws_size,
                              hipStream_t stream) {
    const float* x = (const float*)d_in[0];
    const float* h = (const float*)d_in[1];
    const float* d = (const float*)d_in[2];
    const float* T = (const float*)d_in[3];
    float* out = (float*)d_out;

    int N = in_sizes[0];         // 16*2048*2048 = 67,108,864

    if (ws_size >= 256) {
        int n4    = N >> 2;
        int nfull = n4 / 1024;           // full blocks of 1024 float4s
        int base  = nfull * 4096;        // elements covered by main kernel
        float* cw = (float*)d_ws;

        pack_coefs<<<1, 32, 0, stream>>>(h, d, T, cw);
        if (nfull > 0) {
            // 16384 blocks for the reference shape; covers all of N exactly.
            spike_fwd_pk16<<<nfull, 256, 0, stream>>>(x, cw, out);
        }
        if (base < N) {
            spike_fwd_tail<<<8, 256, 0, stream>>>(x, h, d, T, out, base, N);
        }
    } else {
        // No workspace: scalar fallback over the whole range.
        int blocks = (N + 255) / 256;
        if (blocks > 65536) blocks = 65536;
        if (blocks > 0) {
            spike_fwd_tail<<<blocks, 256, 0, stream>>>(x, h, d, T, out, 0, N);
        }
    }
}